// EncdecProbLossSigmoid_8486855376997
// MI455X (gfx1250) — compile-verified
//
#include <hip/hip_runtime.h>

// MI455X / gfx1250 (CDNA5, wave32). Memory-bound streaming reduction:
//   512 MB logits @ 23.3 TB/s => ~22 us floor. Softplus cost is trimmed to
//   4 VALU + 2 TRANS per element (raw v_exp_f32/v_log_f32, args in safe range)
//   so VALU/TRANS co-execute under the load stream instead of dominating it.
// WMMA f32 16x16x4 with B=ones folds 64 lane-values/instr into the accumulator:
//   D[m][n] = C[m][n] + sum_k A[m][k] => sum(all C) == 16 * sum(all fed A).
// Hot loop has NO bounds check (uniform trip count); the ragged tail is a
// separate uniform-guarded loop with multiply-masking so EXEC stays all-1s
// at every WMMA. Deterministic: no atomics, fixed-slot partials, fixed trees.

typedef __attribute__((ext_vector_type(2))) float v2f;
typedef __attribute__((ext_vector_type(8))) float v8f;

#define NTHR 256
#define MAX_NBLK 2048

__device__ __forceinline__ float softplus_fast(float x) {
  // softplus(x) = max(x,0) + ln2 * log2(1 + exp2(-|x|*log2e))
  // exp2 arg <= 0 (no overflow); log2 arg in (1,2] (no denorm/inf guards needed)
  const float LOG2E = 1.4426950408889634f;
  const float LN2   = 0.6931471805599453f;
  float e = __builtin_amdgcn_exp2f(-fabsf(x) * LOG2E);
  float l = __builtin_amdgcn_logf(1.0f + e);
  return __builtin_fmaf(l, LN2, fmaxf(x, 0.0f));
}

// ---------------- Kernel 1: gather at gt positions (tiny) ----------------
__global__ void __launch_bounds__(1024)
gather_gt_kernel(const float* __restrict__ logits, const int* __restrict__ gt,
                 float* __restrict__ t23, int BS, int V) {
  int tid = threadIdx.x;
  float t2 = 0.0f, t3 = 0.0f;
  for (int i = tid; i < BS; i += 1024) {
    int tok = gt[i];
    float xg = logits[(long long)i * (long long)V + (long long)tok];
    t2 += softplus_fast(-xg);  // = -log_sigmoid(x_gt)
    t3 += softplus_fast(xg);   // = -log_sigmoid(-x_gt)
  }
  __shared__ float s2[1024];
  __shared__ float s3[1024];
  s2[tid] = t2; s3[tid] = t3;
  __syncthreads();
  for (int k = 512; k > 0; k >>= 1) {
    if (tid < k) { s2[tid] += s2[tid + k]; s3[tid] += s3[tid + k]; }
    __syncthreads();
  }
  if (tid == 0) { t23[0] = s2[0]; t23[1] = s3[0]; }
}

// -------- Kernel 2: streaming softplus sum, WMMA-accumulated ----------
__global__ void __launch_bounds__(NTHR)
sp_sum_wmma_kernel(const float* __restrict__ x, float* __restrict__ partials,
                   long long n4) {
  const float4* __restrict__ x4 = reinterpret_cast<const float4*>(x);
  const long long gid     = (long long)blockIdx.x * NTHR + threadIdx.x;
  const long long stride  = (long long)gridDim.x * NTHR;
  const long long stride2 = stride * 2;
  const long long nfull   = n4 / stride2;   // double-iterations with NO bounds check

  v8f c0 = {};
  v8f c1 = {};
  v2f ones; ones[0] = 1.0f; ones[1] = 1.0f;

  // ---- hot loop: unconditional, two b128 loads + two WMMAs per trip ----
  for (long long j = 0; j < nfull; ++j) {
    long long i0 = gid + j * stride2;
    float4 va = x4[i0];
    float4 vb = x4[i0 + stride];
    v2f a0, a1;
    a0[0] = softplus_fast(va.x) + softplus_fast(va.y);
    a0[1] = softplus_fast(va.z) + softplus_fast(va.w);
    a1[0] = softplus_fast(vb.x) + softplus_fast(vb.y);
    a1[1] = softplus_fast(vb.z) + softplus_fast(vb.w);
    c0 = __builtin_amdgcn_wmma_f32_16x16x4_f32(false, a0, false, ones, (short)0, c0, false, false);
    c1 = __builtin_amdgcn_wmma_f32_16x16x4_f32(false, a1, false, ones, (short)0, c1, false, false);
  }

  // ---- ragged tail: <= 2 strided steps, uniform guard, multiply-mask ----
  for (long long base = nfull * stride2; base < n4; base += stride) {
    long long idx = base + gid;
    bool ok = (idx < n4);
    float4 v = {0.0f, 0.0f, 0.0f, 0.0f};
    if (ok) v = x4[idx];
    float m = ok ? 1.0f : 0.0f;          // softplus(0)=ln2, so mask by multiply
    v2f a0;
    a0[0] = m * (softplus_fast(v.x) + softplus_fast(v.y));
    a0[1] = m * (softplus_fast(v.z) + softplus_fast(v.w));
    c0 = __builtin_amdgcn_wmma_f32_16x16x4_f32(false, a0, false, ones, (short)0, c0, false, false);
  }

  // sum of all C entries over the wave == 16 * (sum of all accumulated values)
  float s = (c0[0] + c0[1] + c0[2] + c0[3] + c0[4] + c0[5] + c0[6] + c0[7]) +
            (c1[0] + c1[1] + c1[2] + c1[3] + c1[4] + c1[5] + c1[6] + c1[7]);
  for (int off = 16; off > 0; off >>= 1) s += __shfl_xor(s, off, 32);

  __shared__ float sdata[NTHR / 32];
  int lane = threadIdx.x & 31;
  int wid  = threadIdx.x >> 5;
  if (lane == 0) sdata[wid] = s;
  __syncthreads();
  if (threadIdx.x == 0) {
    float t = 0.0f;
#pragma unroll
    for (int w = 0; w < NTHR / 32; ++w) t += sdata[w];
    partials[blockIdx.x] = t * 0.0625f;   // divide out the x16 column replication
  }
}

// ---------------- Kernel 3: deterministic final combine ----------------
__global__ void __launch_bounds__(NTHR)
finalize_kernel(const float* __restrict__ partials, int nPart,
                const float* __restrict__ t23,
                const float* __restrict__ logits, long long tail_start, int tail_n,
                float* __restrict__ out, double invBS, double invBSV1) {
  int tid = threadIdx.x;
  double s = 0.0;
  for (int i = tid; i < nPart; i += NTHR) s += (double)partials[i];
  __shared__ double sd[NTHR];
  sd[tid] = s;
  __syncthreads();
  for (int k = NTHR / 2; k > 0; k >>= 1) {
    if (tid < k) sd[tid] += sd[tid + k];
    __syncthreads();
  }
  if (tid == 0) {
    double t1 = sd[0];
    for (int i = 0; i < tail_n; ++i)       // scalar tail if total % 4 != 0
      t1 += (double)softplus_fast(logits[tail_start + i]);
    double loss = (double)t23[0] * invBS + (t1 - (double)t23[1]) * invBSV1;
    out[0] = (float)loss;
  }
}

extern "C" void kernel_launch(void* const* d_in, const int* in_sizes, int n_in,
                              void* d_out, int out_size, void* d_ws, size_t ws_size,
                              hipStream_t stream) {
  const float* logits = (const float*)d_in[0];
  const int*   gt     = (const int*)d_in[1];
  float*       out    = (float*)d_out;

  long long total = (long long)in_sizes[0];   // B*S*V
  int BS = in_sizes[1];                       // B*S
  int V  = (int)(total / (long long)BS);

  long long n4 = total >> 2;
  int tail_n = (int)(total & 3);
  long long tail_start = n4 << 2;

  // Workspace: [nblk partial floats][t2][t3]
  int maxPart = (int)(ws_size / sizeof(float)) - 2;
  int nblk = MAX_NBLK;
  if (nblk > maxPart) nblk = maxPart;
  if (nblk < 1) nblk = 1;
  float* partials = (float*)d_ws;
  float* t23      = partials + nblk;

  double invBS   = 1.0 / (double)BS;
  double invBSV1 = 1.0 / ((double)BS * (double)(V - 1));

  gather_gt_kernel<<<1, 1024, 0, stream>>>(logits, gt, t23, BS, V);
  sp_sum_wmma_kernel<<<nblk, NTHR, 0, stream>>>(logits, partials, n4);
  finalize_kernel<<<1, NTHR, 0, stream>>>(partials, nblk, t23,
                                          logits, tail_start, tail_n,
                                          out, invBS, invBSV1);
}